// HGATLayer_64725157151125
// MI455X (gfx1250) — compile-verified
//
#include <hip/hip_runtime.h>
#include <hip/hip_bf16.h>

typedef __attribute__((ext_vector_type(2))) float v2f;
typedef __attribute__((ext_vector_type(8))) float v8f;

#define NODES 50000
#define EDGES 640000
#define DIM   128

// ---------------------------------------------------------------------------
// Zero-fill scratch (atomic accumulation targets must start at 0 every call)
// ---------------------------------------------------------------------------
__global__ void hgat_zero_kernel(float* __restrict__ p, long long n) {
    long long i = (long long)blockIdx.x * blockDim.x + threadIdx.x;
    if (i < n) p[i] = 0.0f;
}

// ---------------------------------------------------------------------------
// C[nrows x 128] = A[nrows x 128] @ W^T[128 x 128] (+ bias), f32 WMMA.
// Block = 256 threads = 8 waves; block covers 16 rows, wave w covers cols
// [16w, 16w+16). K loop: 32 x V_WMMA_F32_16X16X4_F32.
//
// A frag (16x4 f32): lanes 0-15 -> M=lane, K={k,k+1}; lanes 16-31 -> M=lane-16,
// K={k+2,k+3}. B frag (4x16): lanes 0-15 -> N=lane, K={k,k+1}; lanes 16-31 ->
// N=lane-16, K={k+2,k+3}. With B[k][n] = W[n][k] both are aligned float2 loads.
// ---------------------------------------------------------------------------
__global__ void hgat_gemm_wmma(const float* __restrict__ A,
                               const float* __restrict__ W,
                               const float* __restrict__ bias,
                               float* __restrict__ C) {
    const int wave  = threadIdx.x >> 5;
    const int lane  = threadIdx.x & 31;
    const int lm    = lane & 15;          // M (or N) within tile
    const int khalf = (lane >> 4) << 1;   // 0 for lanes 0-15, 2 for lanes 16-31
    const int row0  = blockIdx.x * 16;
    const int n0    = wave * 16;

    const float* arow = A + (long long)(row0 + lm) * DIM;
    const float* wrow = W + (long long)(n0 + lm) * DIM;

    v8f acc = {};
#pragma unroll
    for (int k = 0; k < DIM; k += 4) {
        v2f a = *(const v2f*)(arow + k + khalf);
        v2f b = *(const v2f*)(wrow + k + khalf);
        acc = __builtin_amdgcn_wmma_f32_16x16x4_f32(
            /*neg_a=*/false, a, /*neg_b=*/false, b,
            /*c_mod=*/(short)0, acc, /*reuse_a=*/false, /*reuse_b=*/false);
    }

    const float bv = bias ? bias[n0 + lm] : 0.0f;
    // C/D layout: VGPR r -> M = r (+8 for upper half-lanes), N = lane % 16
    const int mbase = row0 + ((lane >> 4) << 3);
#pragma unroll
    for (int r = 0; r < 8; ++r)
        C[(long long)(mbase + r) * DIM + n0 + lm] = acc[r] + bv;
}

// ---------------------------------------------------------------------------
// One wave32 per edge: gather hr[src], ht[dst] (4 floats/lane), cosine
// attention via wave reduction, scatter a*hr into hsum[dst] with f32 atomics.
// ---------------------------------------------------------------------------
__global__ void hgat_edge_kernel(const float* __restrict__ hr,
                                 const float* __restrict__ ht,
                                 const int* __restrict__ src,
                                 const int* __restrict__ dst,
                                 float* __restrict__ hsum,
                                 float* __restrict__ asum,
                                 float* __restrict__ cnt,
                                 int nedges) {
    const int wavesPerBlock = blockDim.x >> 5;
    const int e = blockIdx.x * wavesPerBlock + (threadIdx.x >> 5);
    if (e >= nedges) return;
    const int lane = threadIdx.x & 31;

    const int s = src[e];
    const int d = dst[e];

    const float4 hv = *(const float4*)(hr + (long long)s * DIM + lane * 4);
    const float4 tv = *(const float4*)(ht + (long long)d * DIM + lane * 4);

    float dot = hv.x * tv.x + hv.y * tv.y + hv.z * tv.z + hv.w * tv.w;
    float nh  = hv.x * hv.x + hv.y * hv.y + hv.z * hv.z + hv.w * hv.w;
    float nt  = tv.x * tv.x + tv.y * tv.y + tv.z * tv.z + tv.w * tv.w;

#pragma unroll
    for (int off = 16; off > 0; off >>= 1) {
        dot += __shfl_xor(dot, off, 32);
        nh  += __shfl_xor(nh,  off, 32);
        nt  += __shfl_xor(nt,  off, 32);
    }

    const float den = fmaxf(sqrtf(nh) * sqrtf(nt), 1e-8f);
    const float a   = dot / den;

    float* hp = hsum + (long long)d * DIM + lane * 4;
    atomicAdd(hp + 0, a * hv.x);
    atomicAdd(hp + 1, a * hv.y);
    atomicAdd(hp + 2, a * hv.z);
    atomicAdd(hp + 3, a * hv.w);
    if (lane == 0) {
        atomicAdd(asum + d, a);
        atomicAdd(cnt  + d, 1.0f);
    }
}

// ---------------------------------------------------------------------------
// Per node: ma_k = a_sum_k / max(cnt_k,1); softmax over the 2 etypes;
// out = w1*h1 + w2*h2. One thread per 4 features.
// ---------------------------------------------------------------------------
__global__ void hgat_combine_kernel(const float* __restrict__ h1,
                                    const float* __restrict__ h2,
                                    const float* __restrict__ a1,
                                    const float* __restrict__ a2,
                                    const float* __restrict__ c1,
                                    const float* __restrict__ c2,
                                    float* __restrict__ out,
                                    int nnodes) {
    const int idx  = blockIdx.x * blockDim.x + threadIdx.x;
    const int node = idx >> 5;
    const int ch   = (idx & 31) * 4;
    if (node >= nnodes) return;

    const float ma1 = a1[node] / fmaxf(c1[node], 1.0f);
    const float ma2 = a2[node] / fmaxf(c2[node], 1.0f);
    const float m   = fmaxf(ma1, ma2);
    const float e1  = __expf(ma1 - m);
    const float e2  = __expf(ma2 - m);
    const float inv = 1.0f / (e1 + e2);
    const float w1  = e1 * inv;
    const float w2  = e2 * inv;

    const long long base = (long long)node * DIM + ch;
    const float4 x1 = *(const float4*)(h1 + base);
    const float4 x2 = *(const float4*)(h2 + base);
    float4 o;
    o.x = w1 * x1.x + w2 * x2.x;
    o.y = w1 * x1.y + w2 * x2.y;
    o.z = w1 * x1.z + w2 * x2.z;
    o.w = w1 * x1.w + w2 * x2.w;
    *(float4*)(out + base) = o;
}

// ---------------------------------------------------------------------------
extern "C" void kernel_launch(void* const* d_in, const int* in_sizes, int n_in,
                              void* d_out, int out_size, void* d_ws, size_t ws_size,
                              hipStream_t stream) {
    const float* feat_vul  = (const float*)d_in[0];
    const float* feat_code = (const float*)d_in[1];
    const int* src[4] = {(const int*)d_in[2], (const int*)d_in[4],
                         (const int*)d_in[6], (const int*)d_in[8]};
    const int* dst[4] = {(const int*)d_in[3], (const int*)d_in[5],
                         (const int*)d_in[7], (const int*)d_in[9]};
    const float* We[4] = {(const float*)d_in[10], (const float*)d_in[11],
                          (const float*)d_in[12], (const float*)d_in[13]};
    const float* W_vul  = (const float*)d_in[14];
    const float* b_vul  = (const float*)d_in[15];
    const float* W_code = (const float*)d_in[16];
    const float* b_code = (const float*)d_in[17];
    float* out = (float*)d_out;

    const long long NF = (long long)NODES * DIM;   // floats per [N,128] array
    float* ws = (float*)d_ws;
    float* ht_vul  = ws;                 // [N,128]
    float* ht_code = ws + NF;            // [N,128]
    float* hr[4]   = {ws + 2 * NF, ws + 3 * NF, ws + 4 * NF, ws + 5 * NF};
    float* hsum[4] = {ws + 6 * NF, ws + 7 * NF, ws + 8 * NF, ws + 9 * NF};
    float* asum[4] = {ws + 10 * NF,             ws + 10 * NF + NODES,
                      ws + 10 * NF + 2 * NODES, ws + 10 * NF + 3 * NODES};
    float* cnt[4]  = {ws + 10 * NF + 4 * NODES, ws + 10 * NF + 5 * NODES,
                      ws + 10 * NF + 6 * NODES, ws + 10 * NF + 7 * NODES};

    // 1) zero the accumulation region (hsum[0..3] + asum[0..3] + cnt[0..3])
    {
        long long nz = 4 * NF + 8 * (long long)NODES;
        int blocks = (int)((nz + 255) / 256);
        hgat_zero_kernel<<<blocks, 256, 0, stream>>>(hsum[0], nz);
    }

    // 2) dense projections via f32 WMMA (50000 rows -> 3125 row tiles of 16)
    const int gblocks = NODES / 16;
    hgat_gemm_wmma<<<gblocks, 256, 0, stream>>>(feat_vul,  W_vul,  b_vul,  ht_vul);
    hgat_gemm_wmma<<<gblocks, 256, 0, stream>>>(feat_code, W_code, b_code, ht_code);
    hgat_gemm_wmma<<<gblocks, 256, 0, stream>>>(feat_code, We[0], nullptr, hr[0]); // e1: code->vul
    hgat_gemm_wmma<<<gblocks, 256, 0, stream>>>(feat_code, We[1], nullptr, hr[1]); // e2: code->vul
    hgat_gemm_wmma<<<gblocks, 256, 0, stream>>>(feat_vul,  We[2], nullptr, hr[2]); // e3: vul->code
    hgat_gemm_wmma<<<gblocks, 256, 0, stream>>>(feat_vul,  We[3], nullptr, hr[3]); // e4: vul->code

    // 3) per-edge cosine attention + segment reductions (one wave per edge)
    const int eblocks = (EDGES + 7) / 8;   // 8 waves / 256-thread block
    const float* htp[4] = {ht_vul, ht_vul, ht_code, ht_code};
    for (int t = 0; t < 4; ++t)
        hgat_edge_kernel<<<eblocks, 256, 0, stream>>>(
            hr[t], htp[t], src[t], dst[t], hsum[t], asum[t], cnt[t], EDGES);

    // 4) stack-softmax combine: out[0]=h_vul (e1,e2), out[1]=h_code (e3,e4)
    const int cblocks = (NODES * 32 + 255) / 256;
    hgat_combine_kernel<<<cblocks, 256, 0, stream>>>(
        hsum[0], hsum[1], asum[0], asum[1], cnt[0], cnt[1], out, NODES);
    hgat_combine_kernel<<<cblocks, 256, 0, stream>>>(
        hsum[2], hsum[3], asum[2], asum[3], cnt[2], cnt[3], out + NF, NODES);
}